// task_19842748907578
// MI455X (gfx1250) — compile-verified
//
#include <hip/hip_runtime.h>
#include <math.h>

#define B_ROWS 8192
#define H_DIM  1024
#define O_DIM  256
#define E_NUM  16

// LDS row stride: 1024 + 4 floats; stride % 64 == 4 so the 32 lanes'
// float2 A-fragment reads touch 32 disjoint bank pairs (no conflicts).
#define LDS_STRIDE 1028

// -------- workspace layout (int32 units) --------
#define EXPERT_OFF 0                      // [B]    expert id per row
#define COUNTS_OFF (B_ROWS)               // [16]   rows per expert
#define PAD_OFF    (B_ROWS + 16)          // [17]   16-padded segment offsets
#define FILL_OFF   (B_ROWS + 33)          // [16]   scatter cursors
#define ROWIDX_OFF (B_ROWS + 64)          // [B+256] grouped row ids (-1 = pad)
#define ROWIDX_N   (B_ROWS + 256)
#define MAX_TILES  (B_ROWS / 16 + E_NUM)  // 528

typedef __attribute__((ext_vector_type(2))) float v2f;
typedef __attribute__((ext_vector_type(8))) float v8f;

__global__ void k_init(int* __restrict__ ws) {
    int i = blockIdx.x * blockDim.x + threadIdx.x;
    if (i < E_NUM) { ws[COUNTS_OFF + i] = 0; ws[FILL_OFF + i] = 0; }
    if (i < ROWIDX_N) ws[ROWIDX_OFF + i] = -1;
}

__global__ void k_expert(const int* __restrict__ num, const int* __restrict__ c,
                         int* __restrict__ ws) {
    int i = blockIdx.x * blockDim.x + threadIdx.x;
    if (i < B_ROWS) {
        int e = c[num[i]];
        ws[EXPERT_OFF + i] = e;
        atomicAdd(&ws[COUNTS_OFF + e], 1);
    }
}

__global__ void k_scan(int* __restrict__ ws) {
    // single thread: 16-entry prefix sum with 16-row padding per expert
    int acc = 0;
    for (int e = 0; e < E_NUM; ++e) {
        ws[PAD_OFF + e] = acc;
        acc += ((ws[COUNTS_OFF + e] + 15) >> 4) << 4;
    }
    ws[PAD_OFF + E_NUM] = acc;
}

__global__ void k_scatter(int* __restrict__ ws) {
    int i = blockIdx.x * blockDim.x + threadIdx.x;
    if (i < B_ROWS) {
        int e = ws[EXPERT_OFF + i];
        int pos = ws[PAD_OFF + e] + atomicAdd(&ws[FILL_OFF + e], 1);
        ws[ROWIDX_OFF + pos] = i;
    }
}

// One block = one 16-row tile of one expert. 8 wave32 x 2 column tiles = N=256.
__global__ __launch_bounds__(256) void k_gemm(const float* __restrict__ x,
                                              const float* __restrict__ W,
                                              const float* __restrict__ bias,
                                              const int* __restrict__ ws,
                                              float* __restrict__ out) {
    __shared__ float As[16 * LDS_STRIDE];
    __shared__ int   sRows[16];
    __shared__ int   sPad[E_NUM + 1];

    const int tid = threadIdx.x;
    const int t   = blockIdx.x;

    if (tid <= E_NUM) sPad[tid] = ws[PAD_OFF + tid];
    __syncthreads();

    const int rowBase = t * 16;
    if (rowBase >= sPad[E_NUM]) return;          // uniform across block

    int e = 0;
    #pragma unroll
    for (int j = 1; j < E_NUM; ++j)
        if (rowBase >= sPad[j]) e = j;           // uniform

    if (tid < 16) sRows[tid] = ws[ROWIDX_OFF + rowBase + tid];
    __syncthreads();

    // ---- stage gathered A rows (16 x 1024 f32) into LDS, float4 coalesced ----
    {
        const int rrow = tid >> 4;               // 0..15: LDS row
        const int rcol = (tid & 15) * 4;         // starting col
        const int srcRow = sRows[rrow];
        const float* __restrict__ xr =
            (srcRow >= 0) ? (x + (size_t)srcRow * H_DIM) : nullptr;
        #pragma unroll
        for (int j = 0; j < 16; ++j) {
            int col = rcol + j * 64;
            float4 v = make_float4(0.f, 0.f, 0.f, 0.f);
            if (xr) v = *(const float4*)(xr + col);
            *(float4*)&As[rrow * LDS_STRIDE + col] = v;
        }
    }
    __syncthreads();

    // ---- WMMA main loop: D(16x16) += A(16x4) x B(4x16), 256 steps over K ----
    const int lane = tid & 31;
    const int wave = tid >> 5;                   // 0..7
    const int half = lane >> 4;                  // K-half select (0: K0/1, 1: K2/3)
    const int l15  = lane & 15;

    const int n0 = wave * 32 + l15;              // column tile 2*wave
    const int n1 = n0 + 16;                      // column tile 2*wave+1
    const float* __restrict__ Wb = W + (size_t)e * O_DIM * H_DIM;
    const float* __restrict__ w0 = Wb + (size_t)n0 * H_DIM + 2 * half;
    const float* __restrict__ w1 = Wb + (size_t)n1 * H_DIM + 2 * half;
    const float* aP = &As[l15 * LDS_STRIDE + 2 * half];

    v8f acc0 = {};
    v8f acc1 = {};
    #pragma unroll 4
    for (int k = 0; k < H_DIM / 4; ++k) {
        v2f a  = *(const v2f*)(aP + 4 * k);      // A frag: x[row=l15, 4k+2h..+1]
        v2f b0 = *(const v2f*)(w0 + 4 * k);      // B frag: W[n0, 4k+2h..+1]
        v2f b1 = *(const v2f*)(w1 + 4 * k);
        acc0 = __builtin_amdgcn_wmma_f32_16x16x4_f32(
            false, a, false, b0, (short)0, acc0, false, false);
        acc1 = __builtin_amdgcn_wmma_f32_16x16x4_f32(
            false, a, false, b1, (short)0, acc1, false, false);
    }

    // ---- epilogue: bias + sigmoid, scatter rows back ----
    const float bias0 = bias[e * O_DIM + n0];
    const float bias1 = bias[e * O_DIM + n1];
    #pragma unroll
    for (int v = 0; v < 8; ++v) {
        const int M = v + 8 * half;              // C/D layout: M = vgpr + 8*half
        const int r = sRows[M];
        if (r >= 0) {
            float z0 = acc0[v] + bias0;
            float z1 = acc1[v] + bias1;
            out[(size_t)r * O_DIM + n0] = 1.f / (1.f + __expf(-z0));
            out[(size_t)r * O_DIM + n1] = 1.f / (1.f + __expf(-z1));
        }
    }
}

extern "C" void kernel_launch(void* const* d_in, const int* in_sizes, int n_in,
                              void* d_out, int out_size, void* d_ws, size_t ws_size,
                              hipStream_t stream) {
    const float* x    = (const float*)d_in[0];   // [B, H]
    const float* W    = (const float*)d_in[1];   // [E, O, H]
    const float* bias = (const float*)d_in[2];   // [E, O]
    const int*   num  = (const int*)d_in[3];     // [B]
    const int*   c    = (const int*)d_in[4];     // [CMAP]
    float*       out  = (float*)d_out;           // [B, O]
    int*         ws   = (int*)d_ws;

    k_init   <<<(ROWIDX_N + 255) / 256, 256, 0, stream>>>(ws);
    k_expert <<<(B_ROWS  + 255) / 256, 256, 0, stream>>>(num, c, ws);
    k_scan   <<<1, 1, 0, stream>>>(ws);
    k_scatter<<<(B_ROWS  + 255) / 256, 256, 0, stream>>>(ws);
    k_gemm   <<<MAX_TILES, 256, 0, stream>>>(x, W, bias, ws, out);
}